// ScanAttention_66288525246579
// MI455X (gfx1250) — compile-verified
//
#include <hip/hip_runtime.h>
#include <hip/hip_bf16.h>
#include <math.h>
#include <stdint.h>

// Problem constants from reference: B=2, H=8, T=1024, P=64.
#define BDIM 2
#define HDIM 8
#define TDIM 1024
#define PDIM 64
#define BH   (BDIM*HDIM)

typedef float v2f __attribute__((ext_vector_type(2)));
typedef float v8f __attribute__((ext_vector_type(8)));

// ---------------------------------------------------------------------------
// Kernel 1: per-(b,h) prefix pass.
//   M[t,p]  = cummax_{s<=t} K[s,p]
//   cE[t,p] = cumsum_{s<=t} exp(2K[s,p])
//   qf[t,p] = exp(Q[t,p] - M[t,p]);  w[t,p] = V*exp(K);  kf[t,p] = exp(K)
// Chunked 2-pass scan: 4 chunks of 256 steps, 64 columns, 256 threads/block.
// ---------------------------------------------------------------------------
__global__ __launch_bounds__(256) void scanattn_prefix_kernel(
    const float* __restrict__ Q, const float* __restrict__ K,
    const float* __restrict__ V,
    float* __restrict__ qf, float* __restrict__ w,
    float* __restrict__ kf, float* __restrict__ cE)
{
    const int bh  = blockIdx.x;
    const int tid = threadIdx.x;
    const int p   = tid & (PDIM - 1);
    const int cz  = tid >> 6;                 // chunk id 0..3
    const int CH  = TDIM / 4;                 // 256 steps per chunk

    __shared__ float mP[4][PDIM];
    __shared__ float eP[4][PDIM];

    const size_t base = (size_t)bh * TDIM * PDIM + p;
    const int tA = cz * CH;

    float mloc = -INFINITY, eloc = 0.f;
    for (int t = tA; t < tA + CH; ++t) {
        float k  = K[base + (size_t)t * PDIM];
        float ek = expf(k);
        mloc = fmaxf(mloc, k);
        eloc += ek * ek;
    }
    mP[cz][p] = mloc;
    eP[cz][p] = eloc;
    __syncthreads();

    float m = -INFINITY, e2 = 0.f;
    for (int c = 0; c < cz; ++c) {
        m  = fmaxf(m, mP[c][p]);
        e2 += eP[c][p];
    }

    for (int t = tA; t < tA + CH; ++t) {
        const size_t idx = base + (size_t)t * PDIM;
        float k = K[idx];
        float q = Q[idx];
        float v = V[idx];
        float ek = expf(k);
        m  = fmaxf(m, k);          // inclusive scan
        e2 += ek * ek;
        qf[idx] = expf(q - m);
        w [idx] = v * ek;
        kf[idx] = ek;
        cE[idx] = e2;
    }
}

// ---------------------------------------------------------------------------
// Kernel 2: causal flash loop, V_WMMA_F32_16X16X4_F32 + double-buffered
// GLOBAL_LOAD_ASYNC_TO_LDS_B128 staging (ASYNCcnt-tracked, overlaps DMA of
// s-tile j+1 with the 32-WMMA chain on s-tile j).
// Grid: (T/64 row tiles, BH heads). Block: 128 threads = 4 wave32.
// ---------------------------------------------------------------------------
#define LDSW 68     // padded row stride (floats) for 16x64 tiles; 272B rows (16B-aligned)
#define SPAD 17     // padded row stride for 16x16 S scratch

#define ASYNC_LOAD_B128(ldsaddr, voff, sbase)                          \
    asm volatile("global_load_async_to_lds_b128 %0, %1, %2"            \
                 :: "v"(ldsaddr), "v"(voff), "s"(sbase) : "memory")

__global__ __launch_bounds__(128) void scanattn_flash_kernel(
    const float* __restrict__ qf, const float* __restrict__ w,
    const float* __restrict__ kf, const float* __restrict__ cE,
    float* __restrict__ out)
{
    __shared__ float w_s [2][16 * LDSW];
    __shared__ float kf_s[2][16 * LDSW];
    __shared__ float s_scr[4 * 16 * SPAD];
    __shared__ float den_s[4 * 16];

    const int bh   = blockIdx.y;
    const int b    = bh >> 3;
    const int h    = bh & 7;
    const int tile = blockIdx.x;
    const int t0   = tile * 64;

    const int tid  = threadIdx.x;
    const int wave = tid >> 5;
    const int lane = tid & 31;
    const int ln   = lane & 15;
    const int lh   = lane >> 4;

    const int rbase = t0 + wave * 16;            // first query row of this wave
    const size_t bbase = (size_t)bh * TDIM * PDIM;

    // Preload A operand (qf rows of this wave) into registers: 16 v2f.
    v2f aq[16];
#pragma unroll
    for (int kc = 0; kc < 16; ++kc)
        aq[kc] = *(const v2f*)&qf[bbase + (size_t)(rbase + ln) * PDIM + 4 * kc + 2 * lh];

    // Denominator: den[t] = sum_p qf[t,p] * cumE2K[t,p]  (one row per lane<16)
    if (lane < 16) {
        const float* qrow = &qf[bbase + (size_t)(rbase + lane) * PDIM];
        const float* crow = &cE[bbase + (size_t)(rbase + lane) * PDIM];
        float s = 0.f;
#pragma unroll 8
        for (int p = 0; p < PDIM; ++p) s += qrow[p] * crow[p];
        den_s[wave * 16 + lane] = s;
    }

    // --- Async staging geometry: each wave DMAs 4 rows of each 16x64 tile. ---
    // 16B chunks: tile = 16 rows x 16 chunks; wave covers 64 chunks via 2 instrs.
    const int chA = wave * 64 + lane;
    const int chB = chA + 32;
    const int rA = chA >> 4, cA = (chA & 15) * 4;
    const int rB = chB >> 4, cB = (chB & 15) * 4;
    const uint32_t gA = (uint32_t)((rA * PDIM + cA) * 4);     // byte off within tile
    const uint32_t gB = (uint32_t)((rB * PDIM + cB) * 4);
    const uint32_t lAo = (uint32_t)((rA * LDSW + cA) * 4);    // byte off within buffer
    const uint32_t lBo = (uint32_t)((rB * LDSW + cB) * 4);
    const float* wsrc = w  + bbase;
    const float* ksrc = kf + bbase;
    // Generic->int truncation yields the raw LDS byte offset (ISA aperture rule).
    const uint32_t wbufa[2] = { (uint32_t)(uintptr_t)&w_s [0][0],
                                (uint32_t)(uintptr_t)&w_s [1][0] };
    const uint32_t kbufa[2] = { (uint32_t)(uintptr_t)&kf_s[0][0],
                                (uint32_t)(uintptr_t)&kf_s[1][0] };

    const v8f vz = {0.f, 0.f, 0.f, 0.f, 0.f, 0.f, 0.f, 0.f};
    v8f acc[4] = {vz, vz, vz, vz};

    const int jmax = tile * 4 + 3;               // s-tiles of 16 up to the diagonal

    // Prologue: DMA s-tile 0 into buffer 0.
    {
        const uint32_t wb = wbufa[0], kb = kbufa[0];
        ASYNC_LOAD_B128(wb + lAo, gA, wsrc);
        ASYNC_LOAD_B128(wb + lBo, gB, wsrc);
        ASYNC_LOAD_B128(kb + lAo, gA, ksrc);
        ASYNC_LOAD_B128(kb + lBo, gB, ksrc);
    }

    for (int j = 0; j <= jmax; ++j) {
        const int s0 = j * 16;

        // Wait for this wave's DMA of tile j, then make all waves' quarters visible.
        asm volatile("s_wait_asynccnt 0x0" ::: "memory");
        __syncthreads();

        // Kick off DMA of tile j+1 into the other buffer (overlaps compute below).
        if (j < jmax) {
            const uint32_t tb = (uint32_t)(j + 1) * (16 * PDIM * 4);
            const uint32_t wb = wbufa[(j + 1) & 1], kb = kbufa[(j + 1) & 1];
            ASYNC_LOAD_B128(wb + lAo, gA + tb, wsrc);
            ASYNC_LOAD_B128(wb + lBo, gB + tb, wsrc);
            ASYNC_LOAD_B128(kb + lAo, gA + tb, ksrc);
            ASYNC_LOAD_B128(kb + lBo, gB + tb, ksrc);
        }

        if (s0 <= rbase + 15) {                  // wave-uniform: skip tiles above diag
            const float* wsb = &w_s [j & 1][0];
            const float* ksb = &kf_s[j & 1][0];

            // ---- GEMM 1: S[m,s] = sum_p qf[m,p] * w[s,p], chained over P=64 ----
            v8f sc = vz;
#pragma unroll
            for (int kc = 0; kc < 16; ++kc) {
                v2f bqk = *(const v2f*)&wsb[ln * LDSW + 4 * kc + 2 * lh];
                sc = __builtin_amdgcn_wmma_f32_16x16x4_f32(
                        false, aq[kc], false, bqk, (short)0, sc, false, false);
            }
            // Causal mask (C layout) — only needed on diagonal-overlapping tiles.
            if (s0 + 15 > rbase) {
#pragma unroll
                for (int v = 0; v < 8; ++v) {
                    const int sg = s0 + ln;
                    const int tg = rbase + v + 8 * lh;
                    if (sg > tg) sc[v] = 0.f;
                }
            }
            // Roundtrip S through LDS to convert C layout -> A layout.
            const int wb2 = wave * 16 * SPAD;
#pragma unroll
            for (int v = 0; v < 8; ++v)
                s_scr[wb2 + (v + 8 * lh) * SPAD + ln] = sc[v];

            // ---- GEMM 2: Y[m,q] += sum_k S[m,s0+k] * kf[s0+k,q] ----
#pragma unroll
            for (int qt = 0; qt < 4; ++qt) {
#pragma unroll
                for (int kc2 = 0; kc2 < 4; ++kc2) {
                    const int k0 = 4 * kc2 + 2 * lh;
                    v2f a2;
                    a2.x = s_scr[wb2 + ln * SPAD + k0];
                    a2.y = s_scr[wb2 + ln * SPAD + k0 + 1];
                    v2f bp;
                    bp.x = ksb[(k0    ) * LDSW + qt * 16 + ln];
                    bp.y = ksb[(k0 + 1) * LDSW + qt * 16 + ln];
                    acc[qt] = __builtin_amdgcn_wmma_f32_16x16x4_f32(
                                 false, a2, false, bp, (short)0, acc[qt], false, false);
                }
            }
        }
    }

    // Epilogue: divide by (den + eps), scatter to (B, T, H*P) layout.
#pragma unroll
    for (int qt = 0; qt < 4; ++qt) {
#pragma unroll
        for (int v = 0; v < 8; ++v) {
            const int t  = rbase + v + 8 * lh;
            const float dn = den_s[wave * 16 + v + 8 * lh];
            const float y  = acc[qt][v] / (dn + 1e-6f);
            out[((size_t)b * TDIM + t) * (HDIM * PDIM) + h * PDIM + qt * 16 + ln] = y;
        }
    }
}

// ---------------------------------------------------------------------------
extern "C" void kernel_launch(void* const* d_in, const int* in_sizes, int n_in,
                              void* d_out, int out_size, void* d_ws, size_t ws_size,
                              hipStream_t stream) {
    const float* Q = (const float*)d_in[0];
    const float* K = (const float*)d_in[1];
    const float* V = (const float*)d_in[2];
    float* out = (float*)d_out;

    float* ws = (float*)d_ws;
    const size_t N = (size_t)BH * TDIM * PDIM;   // 1M floats per buffer
    float* qf = ws;
    float* w  = ws + N;
    float* kf = ws + 2 * N;
    float* cE = ws + 3 * N;                      // 16 MB total workspace

    scanattn_prefix_kernel<<<BH, 256, 0, stream>>>(Q, K, V, qf, w, kf, cE);
    scanattn_flash_kernel<<<dim3(TDIM / 64, BH), 128, 0, stream>>>(qf, w, kf, cE, out);
}